// GIN_37374805410287
// MI455X (gfx1250) — compile-verified
//
#include <hip/hip_runtime.h>
#include <cstddef>

#define N_NODES 100000
#define N_EDGES 1600000
#define DH      128
#define BN_EPS  1e-5f

typedef __attribute__((ext_vector_type(2))) float v2f;
typedef __attribute__((ext_vector_type(8))) float v8f;

// ---------------------------------------------------------------------------
// zero a float buffer
// ---------------------------------------------------------------------------
__global__ void gin_zero_kernel(float* __restrict__ p, size_t n) {
    size_t i = (size_t)blockIdx.x * blockDim.x + threadIdx.x;
    size_t stride = (size_t)gridDim.x * blockDim.x;
    for (; i < n; i += stride) p[i] = 0.0f;
}

// ---------------------------------------------------------------------------
// edge scatter-add: one wave32 per edge, 4 channels per lane
// aggr[dst] += x[src]
// ---------------------------------------------------------------------------
__global__ __launch_bounds__(256) void gin_scatter_kernel(
    const float* __restrict__ x, const int* __restrict__ ei,
    float* __restrict__ aggr)
{
    int lane = threadIdx.x & 31;
    int edge = (int)(((size_t)blockIdx.x * blockDim.x + threadIdx.x) >> 5);
    if (edge >= N_EDGES) return;
    int s = ei[edge];
    int d = ei[N_EDGES + edge];
    const float4 v = ((const float4*)(x + (size_t)s * DH))[lane];
    float* ad = aggr + (size_t)d * DH + lane * 4;
    atomicAdd(ad + 0, v.x);
    atomicAdd(ad + 1, v.y);
    atomicAdd(ad + 2, v.z);
    atomicAdd(ad + 3, v.w);
}

// ---------------------------------------------------------------------------
// fused GEMM via V_WMMA_F32_16X16X4_F32 (wave32)
//   A_eff[n,k] = fuse(A0 (+A1) ; optional BN affine + ReLU)   [n_rows x 128]
//   C = A_eff @ B + bias ; optional ReLU                      [n_rows x n_cols]
// One wave computes a 16x16 tile: 32 chained k-steps (K=128).
// B tile (128 x 16) is held in 64 VGPRs per lane for the whole block lifetime.
// ---------------------------------------------------------------------------
__global__ __launch_bounds__(256) void gin_gemm_wmma_kernel(
    const float* __restrict__ A0, const float* __restrict__ A1,
    const float* __restrict__ scale, const float* __restrict__ shift,
    const float* __restrict__ B, const float* __restrict__ bias,
    float* __restrict__ C, int n_rows, int n_cols, int relu_out)
{
    __shared__ float As[16 * DH];               // 8 KB A tile

    const int lane = threadIdx.x & 31;
    const int wave = threadIdx.x >> 5;          // 0 .. n_cols/16 - 1
    const int m    = lane & 15;                 // row (A) / col (B,C) within tile
    const int kh   = lane >> 4;                 // lane half selects K pair / row half
    const int col  = wave * 16 + m;

    // Preload this wave's B tile: step kk covers K = 4*kk .. 4*kk+3.
    // B (4x16, 32-bit): VGPR0 = K {0 | 2}, VGPR1 = K {1 | 3} split by lane half.
    v2f breg[32];
#pragma unroll
    for (int kk = 0; kk < 32; ++kk) {
        int k0 = kk * 4 + kh * 2;
        breg[kk].x = B[(size_t)(k0 + 0) * n_cols + col];
        breg[kk].y = B[(size_t)(k0 + 1) * n_cols + col];
    }
    const float bval = bias[col];

    const int ntiles = n_rows >> 4;             // 100000/16 = 6250 exact
    for (int tile = blockIdx.x; tile < ntiles; tile += gridDim.x) {
        const int row0 = tile << 4;

        // cooperative fused load of A tile (16 x 128)
        for (int i = threadIdx.x; i < 16 * DH; i += blockDim.x) {
            int r = i >> 7;
            int c = i & (DH - 1);
            size_t idx = (size_t)(row0 + r) * DH + c;
            float v = A0[idx];
            if (A1)   v += A1[idx];
            if (scale) { v = v * scale[c] + shift[c]; v = v > 0.0f ? v : 0.0f; }
            As[i] = v;
        }
        __syncthreads();

        v8f acc;
#pragma unroll
        for (int r = 0; r < 8; ++r) acc[r] = bval;   // bias is per-column

#pragma unroll
        for (int kk = 0; kk < 32; ++kk) {
            int k0 = kk * 4 + kh * 2;
            v2f a;                                   // A 16x4: lane half = K pair
            a.x = As[m * DH + k0 + 0];
            a.y = As[m * DH + k0 + 1];
            acc = __builtin_amdgcn_wmma_f32_16x16x4_f32(
                false, a, false, breg[kk], (short)0, acc, false, false);
        }

        // C/D 16x16: VGPR r -> M = r (lanes 0-15) / M = 8+r (lanes 16-31)
#pragma unroll
        for (int r = 0; r < 8; ++r) {
            int mr = kh ? (8 + r) : r;
            float v = acc[r];
            if (relu_out) v = v > 0.0f ? v : 0.0f;
            C[(size_t)(row0 + mr) * n_cols + col] = v;
        }
        __syncthreads();
    }
}

// ---------------------------------------------------------------------------
// per-channel sum / sumsq over rows (for BatchNorm training-mode stats)
// sums[0..127] = sum, sums[128..255] = sumsq
// ---------------------------------------------------------------------------
__global__ __launch_bounds__(256) void gin_stats_kernel(
    const float* __restrict__ t, float* __restrict__ sums, int n_rows)
{
    __shared__ float ssum[256], ssq[256];
    int c    = threadIdx.x & (DH - 1);
    int half = threadIdx.x >> 7;
    float s = 0.0f, q = 0.0f;
    for (int r = blockIdx.x * 2 + half; r < n_rows; r += gridDim.x * 2) {
        float v = t[(size_t)r * DH + c];
        s += v; q += v * v;
    }
    ssum[threadIdx.x] = s; ssq[threadIdx.x] = q;
    __syncthreads();
    if (threadIdx.x < DH) {
        s = ssum[threadIdx.x] + ssum[threadIdx.x + DH];
        q = ssq[threadIdx.x]  + ssq[threadIdx.x + DH];
        atomicAdd(&sums[c], s);
        atomicAdd(&sums[DH + c], q);
    }
}

// fold BN stats + gamma/beta into per-channel affine: y = x*scale + shift
__global__ void gin_bn_finalize_kernel(
    const float* __restrict__ sums, const float* __restrict__ g,
    const float* __restrict__ be, float* __restrict__ scale,
    float* __restrict__ shift, float inv_n)
{
    int c = threadIdx.x;            // 128 threads
    float mu  = sums[c] * inv_n;
    float var = sums[DH + c] * inv_n - mu * mu;
    float sc  = g[c] * rsqrtf(var + BN_EPS);
    scale[c] = sc;
    shift[c] = be[c] - mu * sc;
}

// ---------------------------------------------------------------------------
extern "C" void kernel_launch(void* const* d_in, const int* in_sizes, int n_in,
                              void* d_out, int out_size, void* d_ws, size_t ws_size,
                              hipStream_t stream) {
    const float* x_in = (const float*)d_in[0];
    const int*   ei   = (const int*)d_in[1];      // JAX demotes int64 -> int32
    const float* W1   = (const float*)d_in[2];
    const float* b1   = (const float*)d_in[3];
    const float* g1   = (const float*)d_in[4];
    const float* be1  = (const float*)d_in[5];
    const float* W2   = (const float*)d_in[6];
    const float* b2   = (const float*)d_in[7];
    const float* fW1  = (const float*)d_in[8];
    const float* fb1  = (const float*)d_in[9];
    const float* fg1  = (const float*)d_in[10];
    const float* fbe1 = (const float*)d_in[11];
    const float* fW2  = (const float*)d_in[12];
    const float* fb2  = (const float*)d_in[13];

    const size_t feat = (size_t)N_NODES * DH;
    float* xb    = (float*)d_ws;            // [N,128] current features
    float* aggr  = xb + feat;               // [N,128] edge aggregation
    float* tb    = aggr + feat;             // [N,128] pre-BN hidden
    float* stats = tb + feat;               // 256: sum | sumsq
    float* scale = stats + 256;             // 128
    float* shift = scale + DH;              // 128

    const int zgrid     = 2048;
    const int gemm_grid = 1280;             // persistent blocks; B tile loaded once
    const int scat_grid = (N_EDGES * 32 + 255) / 256;

    const float* xcur = x_in;
    for (int l = 0; l < 3; ++l) {
        const float* W1l = W1 + (size_t)l * DH * DH;
        const float* W2l = W2 + (size_t)l * DH * DH;

        // aggr = segment_sum(x[src] -> dst)
        gin_zero_kernel<<<zgrid, 256, 0, stream>>>(aggr, feat);
        gin_scatter_kernel<<<scat_grid, 256, 0, stream>>>(xcur, ei, aggr);

        // t = (x + aggr) @ W1 + b1
        gin_gemm_wmma_kernel<<<gemm_grid, 256, 0, stream>>>(
            xcur, aggr, nullptr, nullptr, W1l, b1 + l * DH, tb,
            N_NODES, DH, 0);

        // BN stats over t -> affine
        gin_zero_kernel<<<1, 256, 0, stream>>>(stats, 256);
        gin_stats_kernel<<<1024, 256, 0, stream>>>(tb, stats, N_NODES);
        gin_bn_finalize_kernel<<<1, DH, 0, stream>>>(
            stats, g1 + l * DH, be1 + l * DH, scale, shift, 1.0f / N_NODES);

        // x = relu( relu(BN(t)) @ W2 + b2 )
        gin_gemm_wmma_kernel<<<gemm_grid, 256, 0, stream>>>(
            tb, nullptr, scale, shift, W2l, b2 + l * DH, xb,
            N_NODES, DH, 1);
        xcur = xb;
    }

    // final MLP: t = x @ fW1 + fb1 ; BN ; relu ; out = . @ fW2 + fb2
    gin_gemm_wmma_kernel<<<gemm_grid, 256, 0, stream>>>(
        xcur, nullptr, nullptr, nullptr, fW1, fb1, tb, N_NODES, DH, 0);
    gin_zero_kernel<<<1, 256, 0, stream>>>(stats, 256);
    gin_stats_kernel<<<1024, 256, 0, stream>>>(tb, stats, N_NODES);
    gin_bn_finalize_kernel<<<1, DH, 0, stream>>>(
        stats, fg1, fbe1, scale, shift, 1.0f / N_NODES);
    gin_gemm_wmma_kernel<<<gemm_grid, 128, 0, stream>>>(
        tb, nullptr, scale, shift, fW2, fb2, (float*)d_out, N_NODES, 64, 0);
}